// BandSplit_51539607553038
// MI455X (gfx1250) — compile-verified
//
#include <hip/hip_runtime.h>

typedef float v2f __attribute__((ext_vector_type(2)));
typedef float v8f __attribute__((ext_vector_type(8)));

#define NBANDS 31
#define BATCH  16
#define FREQ   257
#define TLEN   1000
#define CH     128
#define MAXC   34

__constant__ int c_band[NBANDS] = {2,3,3,3,3,3,3,3,3,3,3,8,8,8,8,8,8,8,8,8,8,8,8,16,16,16,16,16,16,16,17};
__constant__ int c_f0[NBANDS]   = {0,2,5,8,11,14,17,20,23,26,29,32,40,48,56,64,72,80,88,96,104,112,120,128,144,160,176,192,208,224,240};

// ---------------------------------------------------------------------------
// Pass 1: per (band, batch) mean / rstd over the contiguous slice
// x[bb, f0:f0+b, :, :]  (b * T * 2 contiguous floats)
// stats[(band*16+bb)*2 + {0,1}] = {mu, rstd}
// ---------------------------------------------------------------------------
__global__ __launch_bounds__(256) void band_stats(const float* __restrict__ x,
                                                  float* __restrict__ stats) {
  const int g    = blockIdx.x;          // 0..495
  const int band = g >> 4;
  const int bb   = g & 15;
  const int b    = c_band[band];
  const int f0   = c_f0[band];
  const int n    = b * (TLEN * 2);
  const float* p = x + ((long)bb * FREQ + f0) * (TLEN * 2);

  float s = 0.f, s2 = 0.f;
  for (int i = threadIdx.x; i < n; i += 256) {
    float v = p[i];
    s += v;
    s2 += v * v;
  }
  __shared__ float sh[256], sh2[256];
  sh[threadIdx.x]  = s;
  sh2[threadIdx.x] = s2;
  __syncthreads();
  for (int off = 128; off > 0; off >>= 1) {
    if ((int)threadIdx.x < off) {
      sh[threadIdx.x]  += sh[threadIdx.x + off];
      sh2[threadIdx.x] += sh2[threadIdx.x + off];
    }
    __syncthreads();
  }
  if (threadIdx.x == 0) {
    float inv = 1.0f / (float)n;
    float mu  = sh[0] * inv;
    float var = sh2[0] * inv - mu * mu;
    stats[g * 2 + 0] = mu;
    stats[g * 2 + 1] = rsqrtf(var + 1e-5f);
  }
}

// ---------------------------------------------------------------------------
// Pass 2: fused GroupNorm-affine + Linear(c->128) via V_WMMA_F32_16X16X4_F32.
// Grid (125, 31), 256 threads = 8 waves. Wave w owns M-tile (blockIdx.x*8+w)
// of 16 (batch,t) rows and all 128 output channels (8 N-tiles of 16).
//
// k-loop is split into c>>2 fully-uniform steps (no guards, no exec games)
// plus one branchless masked tail step iff (c & 2): both lane halves read
// channel pair (c-2, c-1); half==1 lanes multiply their fragments by 0.0f.
//
// A fragment (16x4 f32, 2 VGPRs): lanes 0-15 K=0,1 ; lanes 16-31 K=2,3.
// B fragment (4x16  f32, 2 VGPRs): same K split, lanes index N.
// C/D (16x16 f32, 8 VGPRs): vgpr r holds M=r (lanes<16) / M=r+8 (lanes>=16).
// ---------------------------------------------------------------------------
__global__ __launch_bounds__(256) void band_gemm(const float* __restrict__ x,
                                                 const float* __restrict__ gn_w,
                                                 const float* __restrict__ gn_b,
                                                 const float* __restrict__ fc_w,
                                                 const float* __restrict__ fc_b,
                                                 const float* __restrict__ stats,
                                                 float* __restrict__ out) {
  const int band = blockIdx.y;
  const int c    = 2 * c_band[band];
  const int f0   = c_f0[band];
  const int lane = threadIdx.x & 31;
  const int wave = threadIdx.x >> 5;
  const int base = (blockIdx.x * 8 + wave) * 16;   // first row of this M tile
  const int half = lane >> 4;                      // 0: K lo / M 0-7 ; 1: K hi / M 8-15
  const int l15  = lane & 15;

  // Row this lane feeds into the A fragment (M = lane % 16).
  const int rowA = base + l15;
  const int bbA  = rowA / TLEN;
  const int tA   = rowA - bbA * TLEN;
  const float mu   = stats[(band * 16 + bbA) * 2 + 0];
  const float rstd = stats[(band * 16 + bbA) * 2 + 1];

  const float* gw = gn_w + band * MAXC;
  const float* gb = gn_b + band * MAXC;
  const float* W  = fc_w + (long)band * CH * MAXC;

  v8f acc[8] = {};

  // ---- full k-steps: k0 = 4*kk + 2*half always < c, fully uniform ----
  const int kfull = c >> 2;
  // per-lane base pointers; advance by fixed strides each iteration
  const float* xp = x + (((long)bbA * FREQ + f0 + half) * TLEN + tA) * 2;
  const float* wp = W + l15 * MAXC + 2 * half;

  for (int kk = 0; kk < kfull; ++kk) {
    const int k0 = 4 * kk + 2 * half;
    const float vx = xp[0];
    const float vy = xp[1];
    v2f a;
    a.x = (vx - mu) * rstd * gw[k0]     + gb[k0];
    a.y = (vy - mu) * rstd * gw[k0 + 1] + gb[k0 + 1];
#pragma unroll
    for (int nt = 0; nt < 8; ++nt) {
      v2f bf;
      bf.x = wp[nt * 16 * MAXC + 0];
      bf.y = wp[nt * 16 * MAXC + 1];
      acc[nt] = __builtin_amdgcn_wmma_f32_16x16x4_f32(
          false, a, false, bf, (short)0, acc[nt], false, false);
    }
    xp += 2 * TLEN * 2;   // freq advances by 2 per k-step
    wp += 4;              // k advances by 4
  }

  // ---- masked tail step (c % 4 == 2): channels (c-2, c-1) ----
  if (c & 2) {
    const float m  = half ? 0.0f : 1.0f;
    const int  kt  = c - 2;
    const float* xt = x + (((long)bbA * FREQ + f0 + (kt >> 1)) * TLEN + tA) * 2;
    const float vx = xt[0];
    const float vy = xt[1];
    v2f a;
    a.x = m * ((vx - mu) * rstd * gw[kt]     + gb[kt]);
    a.y = m * ((vy - mu) * rstd * gw[kt + 1] + gb[kt + 1]);
    const float* wt = W + l15 * MAXC + kt;
#pragma unroll
    for (int nt = 0; nt < 8; ++nt) {
      v2f bf;
      bf.x = m * wt[nt * 16 * MAXC + 0];
      bf.y = m * wt[nt * 16 * MAXC + 1];
      acc[nt] = __builtin_amdgcn_wmma_f32_16x16x4_f32(
          false, a, false, bf, (short)0, acc[nt], false, false);
    }
  }

  // ---- epilogue: add fc_b, store to z[b, t, o, band] (stride NBANDS in o) ----
  const int rowc = base + 8 * half;     // C rows for this lane half
#pragma unroll
  for (int nt = 0; nt < 8; ++nt) {
    const int n = nt * 16 + l15;
    const float bias = fc_b[band * CH + n];
#pragma unroll
    for (int r = 0; r < 8; ++r) {
      const int row = rowc + r;
      out[(row * CH + n) * NBANDS + band] = acc[nt][r] + bias;
    }
  }
}

extern "C" void kernel_launch(void* const* d_in, const int* in_sizes, int n_in,
                              void* d_out, int out_size, void* d_ws, size_t ws_size,
                              hipStream_t stream) {
  const float* x    = (const float*)d_in[0];
  const float* gn_w = (const float*)d_in[1];
  const float* gn_b = (const float*)d_in[2];
  const float* fc_w = (const float*)d_in[3];
  const float* fc_b = (const float*)d_in[4];
  float* out   = (float*)d_out;
  float* stats = (float*)d_ws;   // 496 * 2 floats

  band_stats<<<dim3(NBANDS * BATCH), dim3(256), 0, stream>>>(x, stats);
  band_gemm<<<dim3(125, NBANDS), dim3(256), 0, stream>>>(x, gn_w, gn_b, fc_w,
                                                         fc_b, stats, out);
}